// ThermoQuantizer_50122268345057
// MI455X (gfx1250) — compile-verified
//
#include <hip/hip_runtime.h>
#include <stdint.h>

#define WAVES  8      // 256 threads per block
#define NITER  4      // groups per wave per block (double-buffered pipeline)
#define GROUP  128    // elements per group (= 32 lanes x float4)

// One async global->LDS copy: 16 bytes per lane (512B per wave = one group).
// GVS mode: 64-bit SGPR base + 32-bit per-lane byte offset; LDS dest byte offset
// in VGPR (VDST field). Tracked by ASYNCcnt.
__device__ __forceinline__ void async_copy_b128(const float* base, uint32_t goff, uint32_t loff) {
    asm volatile("global_load_async_to_lds_b128 %0, %1, %2 offset:0"
                 :: "v"(loff), "v"(goff), "s"(base)
                 : "memory");
}

__global__ __launch_bounds__(WAVES * 32)
void thermo_quant_kernel(const float* __restrict__ x,
                         const float* __restrict__ cb_g,
                         const float* __restrict__ p_pressure,
                         const float* __restrict__ p_temp,
                         float* __restrict__ out,
                         int ngroups)
{
    __shared__ __align__(16) float lds[2][WAVES][GROUP];

    const int lane = threadIdx.x & 31;
    const int wave = threadIdx.x >> 5;

    // ---- uniform constants (uniform addresses -> SMEM loads) ----
    const float pressure = p_pressure[0];
    const float temp     = p_temp[0];
    const float k        = 1.44269504088896340736f / (temp + 1e-6f); // log2(e)/T
    float cb[16], a[16], b[16];
#pragma unroll
    for (int c = 0; c < 16; ++c) {
        cb[c] = cb_g[c];
        a[c]  = 2.0f * cb[c] * k;   // logit/T in log2 domain = fma(xn, a, b); xn^2 cancels
        b[c]  = -cb[c] * cb[c] * k;
    }

    // LDS byte offset of shared buffer: low 32 bits of generic pointer
    // (ISA aperture rule).
    const uint32_t lds_base = (uint32_t)(uintptr_t)&lds[0][0][0];

    const int gbase = blockIdx.x * (WAVES * NITER);

    // gs: wave-uniform group index forced into an SGPR so guards become scalar
    // branches (no EXEC-mask divergence) and addressing stays saddr-based.
    auto issue = [&](int gs, int buf) {
        const uint32_t goff = (uint32_t)gs * (GROUP * 4u) + (uint32_t)lane * 16u;
        const uint32_t loff = lds_base + (uint32_t)(buf * WAVES + wave) * (GROUP * 4u)
                                       + (uint32_t)lane * 16u;
        async_copy_b128(x, goff, loff);
    };

    auto compute = [&](int gs, int buf) {
        const float4 v = *(const float4*)&lds[buf][wave][lane * 4];

        // group abs-mean via intra-wave reduction (wave32 == one group)
        float s = fabsf(v.x) + fabsf(v.y) + fabsf(v.z) + fabsf(v.w);
#pragma unroll
        for (int m = 16; m >= 1; m >>= 1) s += __shfl_xor(s, m, 32);
        const float meanc = fmaxf(s * (1.0f / GROUP), 1e-5f); // clip(mean, 1e-5)
        const float scale = 1.0f / meanc;

        const float xe[4] = {v.x, v.y, v.z, v.w};
        float r[4];
#pragma unroll
        for (int e = 0; e < 4; ++e) {
            const float xn = xe[e] * scale;
            float t[16], m = -3.0e38f;
#pragma unroll
            for (int c = 0; c < 16; ++c) {
                t[c] = fmaf(xn, a[c], b[c]);
                m = fmaxf(m, t[c]);
            }
            float se = 0.0f, sw = 0.0f;
#pragma unroll
            for (int c = 0; c < 16; ++c) {
                const float p = exp2f(t[c] - m);   // native v_exp_f32
                se += p;
                sw = fmaf(p, cb[c], sw);
            }
            const float qx = (sw / se) * meanc;    // qx_norm / scale
            r[e] = fmaf(pressure, qx - xe[e], xe[e]);
        }
        float4 o = make_float4(r[0], r[1], r[2], r[3]);
        *(float4*)(out + (size_t)gs * GROUP + lane * 4) = o;
    };

    // ---- double-buffered async pipeline (per-wave ASYNCcnt, in-order completion) ----
    {
        const int g0 = __builtin_amdgcn_readfirstlane(gbase + wave);
        if (g0 < ngroups) issue(g0, 0);
    }
#pragma unroll
    for (int i = 0; i < NITER; ++i) {
        const int g     = __builtin_amdgcn_readfirstlane(gbase + i * WAVES + wave);
        const int gnext = __builtin_amdgcn_readfirstlane(gbase + (i + 1) * WAVES + wave);
        if ((i + 1 < NITER) && (gnext < ngroups)) {
            issue(gnext, (i + 1) & 1);
            asm volatile("s_wait_asynccnt 1" ::: "memory"); // buffer i landed, next in flight
        } else {
            asm volatile("s_wait_asynccnt 0" ::: "memory");
        }
        if (g < ngroups) compute(g, i & 1);
    }
}

extern "C" void kernel_launch(void* const* d_in, const int* in_sizes, int n_in,
                              void* d_out, int out_size, void* d_ws, size_t ws_size,
                              hipStream_t stream) {
    const float* x        = (const float*)d_in[0];
    const float* codebook = (const float*)d_in[1];
    const float* pressure = (const float*)d_in[2];
    const float* temp     = (const float*)d_in[3];
    float* out            = (float*)d_out;

    const int n       = in_sizes[0];
    const int ngroups = n / GROUP;
    const int gpb     = WAVES * NITER;
    const int blocks  = (ngroups + gpb - 1) / gpb;

    thermo_quant_kernel<<<blocks, WAVES * 32, 0, stream>>>(
        x, codebook, pressure, temp, out, ngroups);
}